// Model_54365696033600
// MI455X (gfx1250) — compile-verified
//
#include <hip/hip_runtime.h>
#include <hip/hip_bf16.h>
#include <math.h>

typedef __attribute__((ext_vector_type(2))) float v2f;
typedef __attribute__((ext_vector_type(8))) float v8f;

#define MOM 0.9f
#define BN_EPS 1e-5f
#define NORM_EPS 1e-12f

// ---------------------------------------------------------------------------
// wave32 helpers
// ---------------------------------------------------------------------------
__device__ __forceinline__ float wave_reduce_add(float s) {
#pragma unroll
  for (int off = 16; off >= 1; off >>= 1) s += __shfl_xor(s, off, 32);
  return s;
}

// ---------------------------------------------------------------------------
// Register-blocked fp32 GEMM via V_WMMA_F32_16X16X4_F32.
//   C[M,Nreal] = A[M,K] * B (+ optional per-column bias)
//   BT = false: B row-major [K,N] (ldb = N)
//   BT = true : B element (k,n) = B[n*ldb + k] (B given as [N,K] rows; rows
//               up to Npad must be readable — caller pads the buffer)
//
// Each wave owns a 64x32 macro-tile = 4(M) x 2(N) WMMA tiles:
//   - 8 accumulators (64 VGPRs) — fits the allocator's budget with zero
//     spills (4x4 / 128 acc VGPRs spilled accumulators to scratch)
//   - per K-step of 4: 4 A frags + 2 B frags feed 8 WMMAs (0.75 loads/WMMA)
//   - A frags reused across N-tiles, B frags across M-tiles (registers only)
// 8 waves / 256-thread block; macro-tiles flattened across the grid.
// Fragment layout per CDNA5 ISA 7.12.2:
//   A 16x4 : lane half h, lane&15 = M row; VGPR0/1 = {K=2h, K=2h+1}
//   B 4x16 : mirrored (VGPR0/1 = rows K=2h, K=2h+1; N = lane&15)
//   C 16x16: VGPR r -> M = r + 8h, N = lane&15
// ---------------------------------------------------------------------------
template <bool BT>
__global__ void gemm_wmma_f32(const float* __restrict__ A,
                              const float* __restrict__ B,
                              const float* __restrict__ bias,
                              float* __restrict__ C,
                              int M, int Npad, int Nreal, int K,
                              int lda, int ldb, int ldc) {
  const int wave = threadIdx.x >> 5;
  const int lane = threadIdx.x & 31;
  const int half = lane >> 4;
  const int lm   = lane & 15;

  const int ngrp  = Npad >> 5;            // 32-wide N groups
  const int total = (M >> 6) * ngrp;      // 64-tall M groups
  const int g     = blockIdx.x * (blockDim.x >> 5) + wave;
  if (g >= total) return;                 // wave-uniform: EXEC all-ones

  const int m0 = (g / ngrp) << 6;
  const int n0 = (g % ngrp) << 5;

  v8f acc[4][2];
#pragma unroll
  for (int i = 0; i < 4; ++i)
#pragma unroll
    for (int j = 0; j < 2; ++j) acc[i][j] = (v8f){};

  // Per-fragment base pointers (K offset added in the loop).
  const float* pa[4];
#pragma unroll
  for (int i = 0; i < 4; ++i)
    pa[i] = A + (size_t)(m0 + 16 * i + lm) * lda + 2 * half;

  if (BT) {
    const float* pb[2];
#pragma unroll
    for (int j = 0; j < 2; ++j)
      pb[j] = B + (size_t)(n0 + 16 * j + lm) * ldb + 2 * half;

#pragma unroll 2
    for (int k = 0; k < K; k += 4) {
      v2f a[4], b[2];
#pragma unroll
      for (int i = 0; i < 4; ++i) a[i] = *(const v2f*)(pa[i] + k);
#pragma unroll
      for (int j = 0; j < 2; ++j) b[j] = *(const v2f*)(pb[j] + k);
#pragma unroll
      for (int i = 0; i < 4; ++i)
#pragma unroll
        for (int j = 0; j < 2; ++j)
          acc[i][j] = __builtin_amdgcn_wmma_f32_16x16x4_f32(
              false, a[i], false, b[j], (short)0, acc[i][j], false, false);
    }
  } else {
    const float* pb[2];
#pragma unroll
    for (int j = 0; j < 2; ++j)
      pb[j] = B + (size_t)(2 * half) * ldb + (n0 + 16 * j + lm);

#pragma unroll 2
    for (int k = 0; k < K; k += 4) {
      v2f a[4], b[2];
#pragma unroll
      for (int i = 0; i < 4; ++i) a[i] = *(const v2f*)(pa[i] + k);
#pragma unroll
      for (int j = 0; j < 2; ++j) {
        b[j].x = pb[j][(size_t)k * ldb];
        b[j].y = pb[j][(size_t)(k + 1) * ldb];
      }
#pragma unroll
      for (int i = 0; i < 4; ++i)
#pragma unroll
        for (int j = 0; j < 2; ++j)
          acc[i][j] = __builtin_amdgcn_wmma_f32_16x16x4_f32(
              false, a[i], false, b[j], (short)0, acc[i][j], false, false);
    }
  }

  // Stores (divergent column guard is fine: all WMMAs already issued).
#pragma unroll
  for (int j = 0; j < 2; ++j) {
    const int col = n0 + 16 * j + lm;
    if (col >= Nreal) continue;
    const float bv = bias ? bias[col] : 0.0f;
#pragma unroll
    for (int i = 0; i < 4; ++i) {
#pragma unroll
      for (int r = 0; r < 8; ++r) {
        const int row = m0 + 16 * i + r + 8 * half;
        C[(size_t)row * ldc + col] = acc[i][j][r] + bv;
      }
    }
  }
}

// ---------------------------------------------------------------------------
// Per-column batch statistics (BatchNorm1d training mode, biased variance).
// ---------------------------------------------------------------------------
__global__ void col_stats(const float* __restrict__ h,
                          float* __restrict__ mu, float* __restrict__ rstd,
                          int rows, int cols) {
  const int c = blockIdx.x;
  float s = 0.0f, s2 = 0.0f;
  for (int r = threadIdx.x; r < rows; r += blockDim.x) {
    const float v = h[(size_t)r * cols + c];
    s += v;
    s2 += v * v;
  }
  __shared__ float sh1[8], sh2[8];
  s  = wave_reduce_add(s);
  s2 = wave_reduce_add(s2);
  if ((threadIdx.x & 31) == 0) {
    sh1[threadIdx.x >> 5] = s;
    sh2[threadIdx.x >> 5] = s2;
  }
  __syncthreads();
  if (threadIdx.x == 0) {
    float t1 = 0.0f, t2 = 0.0f;
#pragma unroll
    for (int i = 0; i < 8; ++i) { t1 += sh1[i]; t2 += sh2[i]; }
    const float m   = t1 / (float)rows;
    const float var = t2 / (float)rows - m * m;
    mu[c]   = m;
    rstd[c] = 1.0f / sqrtf(var + BN_EPS);
  }
}

// ---------------------------------------------------------------------------
// In-place BN affine + ReLU over [rows, cols].
// ---------------------------------------------------------------------------
__global__ void bn_relu(float* __restrict__ h,
                        const float* __restrict__ mu,
                        const float* __restrict__ rstd,
                        const float* __restrict__ gamma,
                        const float* __restrict__ beta,
                        int n, int cols) {
  const int i = blockIdx.x * blockDim.x + threadIdx.x;
  if (i >= n) return;
  const int c = i % cols;
  const float v = (h[i] - mu[c]) * rstd[c] * gamma[c] + beta[c];
  h[i] = fmaxf(v, 0.0f);
}

// ---------------------------------------------------------------------------
// In-place row L2 normalization: x / max(||x||, 1e-12).
// ---------------------------------------------------------------------------
__global__ void row_l2norm(float* __restrict__ x, int cols) {
  const int row = blockIdx.x;
  float* p = x + (size_t)row * cols;
  const float v = p[threadIdx.x];
  float s = wave_reduce_add(v * v);
  __shared__ float sh[8];
  __shared__ float scale_sh;
  if ((threadIdx.x & 31) == 0) sh[threadIdx.x >> 5] = s;
  __syncthreads();
  if (threadIdx.x == 0) {
    float t = 0.0f;
#pragma unroll
    for (int i = 0; i < 8; ++i) t += sh[i];
    scale_sh = 1.0f / fmaxf(sqrtf(t), NORM_EPS);
  }
  __syncthreads();
  p[threadIdx.x] = v * scale_sh;
}

// ---------------------------------------------------------------------------
// Zero fill (for the padded tail of the memory-bank workspace copy).
// ---------------------------------------------------------------------------
__global__ void zero_fill(float* __restrict__ p, int n) {
  const int i = blockIdx.x * blockDim.x + threadIdx.x;
  if (i < n) p[i] = 0.0f;
}

// ---------------------------------------------------------------------------
// Strictly sequential momentum memory update (order matters for duplicate
// labels). Single wave32; 8 contiguous floats per lane, cross-lane sum via
// __shfl_xor. Same-wave store->load to the same address is kept in order by
// the memory system (CDNA5 ISA 7.3 VMEM ordering).
// ---------------------------------------------------------------------------
__global__ void mem_update(float* __restrict__ mem,
                           const float* __restrict__ feat,
                           const int* __restrict__ label, int n) {
  const int lane = threadIdx.x;  // 0..31
  for (int i = 0; i < n; ++i) {
    const int y = label[i];
    float* mr = mem + (size_t)y * 256 + lane * 8;
    const float* fr = feat + (size_t)i * 256 + lane * 8;
    float p[8];
    float s = 0.0f;
#pragma unroll
    for (int j = 0; j < 8; ++j) {
      const float pv = MOM * mr[j] + (1.0f - MOM) * fr[j];
      p[j] = pv;
      s += pv * pv;
    }
#pragma unroll
    for (int off = 16; off >= 1; off >>= 1) s += __shfl_xor(s, off, 32);
    const float r = 1.0f / fmaxf(sqrtf(s), NORM_EPS);
#pragma unroll
    for (int j = 0; j < 8; ++j) mr[j] = p[j] * r;
  }
}

// ---------------------------------------------------------------------------
// Driver
// ---------------------------------------------------------------------------
extern "C" void kernel_launch(void* const* d_in, const int* in_sizes, int n_in,
                              void* d_out, int out_size, void* d_ws, size_t ws_size,
                              hipStream_t stream) {
  (void)in_sizes; (void)n_in; (void)out_size; (void)ws_size;

  const float* feat_in   = (const float*)d_in[0];  // [4096, 1536]
  const int*   label     = (const int*)  d_in[1];  // [4096]
  const float* W1        = (const float*)d_in[2];  // [1536, 512]
  const float* b1        = (const float*)d_in[3];  // [512]
  const float* gamma     = (const float*)d_in[4];  // [512]
  const float* beta      = (const float*)d_in[5];  // [512]
  const float* W2        = (const float*)d_in[6];  // [512, 256]
  const float* b2        = (const float*)d_in[7];  // [256]
  const float* memory_pt = (const float*)d_in[8];  // [10000, 256]
  float* out = (float*)d_out;                      // [4096, 10000]

  const int N = 4096, DH = 512, DIN = 1536, DF = 256, NC = 10000;
  const int NCPAD = 10016;  // 626 tiles of 16 -> 313 groups of 32

  // workspace layout (floats)
  float* ws   = (float*)d_ws;
  float* h    = ws;                       // 4096*512
  float* mu   = h + (size_t)N * DH;       // 512
  float* rstd = mu + DH;                  // 512
  float* feat = rstd + DH;                // 4096*256
  float* mem  = feat + (size_t)N * DF;    // 10016*256 (padded)

  // 1) h = feat_in @ W1 + b1     groups = 64 * (512/32)=16 -> 1024 waves
  gemm_wmma_f32<false><<<128, 256, 0, stream>>>(
      feat_in, W1, b1, h, N, DH, DH, DIN, DIN, DH, DH);

  // 2) batch statistics per column
  col_stats<<<DH, 256, 0, stream>>>(h, mu, rstd, N, DH);

  // 3) BN + ReLU in place
  bn_relu<<<(N * DH) / 256, 256, 0, stream>>>(h, mu, rstd, gamma, beta,
                                              N * DH, DH);

  // 4) feat = h @ W2 + b2        groups = 64 * 8 -> 512 waves
  gemm_wmma_f32<false><<<64, 256, 0, stream>>>(
      h, W2, b2, feat, N, DF, DF, DH, DH, DF, DF);

  // 5) row-wise L2 normalize in place
  row_l2norm<<<N, 256, 0, stream>>>(feat, DF);

  // 6) private padded copy of the memory bank (inputs stay untouched)
  hipMemcpyAsync(mem, memory_pt, sizeof(float) * (size_t)NC * DF,
                 hipMemcpyDeviceToDevice, stream);
  zero_fill<<<((NCPAD - NC) * DF) / 256, 256, 0, stream>>>(
      mem + (size_t)NC * DF, (NCPAD - NC) * DF);

  // 7) sequential momentum + renorm scan
  mem_update<<<1, 32, 0, stream>>>(mem, feat, label, N);

  // 8) sim = feat @ mem^T        groups = 64 * 313 = 20032 waves -> 2504 blocks
  gemm_wmma_f32<true><<<2504, 256, 0, stream>>>(
      feat, mem, nullptr, out, N, NCPAD, NC, DF, DF, DF, NC);
}